// LowRankInterpLinearOperator2d_29171417874661
// MI455X (gfx1250) — compile-verified
//
#include <hip/hip_runtime.h>

typedef float v2f __attribute__((ext_vector_type(2)));
typedef float v8f __attribute__((ext_vector_type(8)));

#define CIN  128
#define COUT 128
#define HWD  128           // H == W == 128
#define LDSTRIDE 136       // padded v-tile row stride (bank-conflict-free B frags)

// align_corners-style bilinear weights for a 4-point axis sampled at (idx+0.5)/128*3
__device__ __forceinline__ void interp_w(int idx, float w[4]) {
    float t  = ((float)idx + 0.5f) * (3.0f / 128.0f);
    float fi = floorf(t);
    fi = fminf(fmaxf(fi, 0.0f), 2.0f);
    float f  = t - fi;
    int i0 = (int)fi;
#pragma unroll
    for (int p = 0; p < 4; ++p)
        w[p] = (p == i0) ? (1.0f - f) : ((p == i0 + 1) ? f : 0.0f);
}

// ---------------------------------------------------------------------------
// Kernel 0: pre-swizzle conv_w into WMMA A-fragment layout; fold conv_b+bias.
// Aswz[((ot*32+ks)*32+lane)*2 + {0,1}] = conv_w[ot*16 + (lane&15)][ks*4 + ((lane>>4)*2) + {0,1}]
// ---------------------------------------------------------------------------
__global__ void prep_kernel(const float* __restrict__ conv_w,
                            const float* __restrict__ conv_b,
                            const float* __restrict__ bias,
                            float* __restrict__ Aswz,
                            float* __restrict__ cbb) {
    int tid = blockIdx.x * 256 + threadIdx.x;
    if (tid < 8192) {
        int lane = tid & 31;
        int ks   = (tid >> 5) & 31;
        int ot   = tid >> 10;
        int mrow = lane & 15;
        int krow = ks * 4 + ((lane >> 4) << 1);
        const float* src = conv_w + (ot * 16 + mrow) * CIN + krow;
        Aswz[tid * 2 + 0] = src[0];
        Aswz[tid * 2 + 1] = src[1];
    } else if (tid < 8192 + COUT) {
        int o = tid - 8192;
        cbb[o] = conv_b[o] + bias[o];
    }
}

// ---------------------------------------------------------------------------
// Kernel 1: vproj[b,i,p,q] = sum_{h,w} wy[h,p]*wx[w,q]*v[b,i,h,w]
// one block per (b,i) image; separable: reduce over w first (4 accs), scale by wy.
// ---------------------------------------------------------------------------
__global__ void vproj_kernel(const float* __restrict__ v, float* __restrict__ vproj) {
    __shared__ float red[256 * 17];
    int t = threadIdx.x;
    int h = t >> 1, half = t & 1;
    float wy[4]; interp_w(h, wy);

    const float4* g = (const float4*)(v + (size_t)blockIdx.x * (HWD * HWD) + h * HWD + half * 64);
    float rq0 = 0.f, rq1 = 0.f, rq2 = 0.f, rq3 = 0.f;
#pragma unroll 4
    for (int j = 0; j < 16; ++j) {
        float4 x = g[j];
        int w0 = half * 64 + j * 4;
        float e[4] = {x.x, x.y, x.z, x.w};
#pragma unroll
        for (int k = 0; k < 4; ++k) {
            float wx[4]; interp_w(w0 + k, wx);
            rq0 += wx[0] * e[k]; rq1 += wx[1] * e[k];
            rq2 += wx[2] * e[k]; rq3 += wx[3] * e[k];
        }
    }
    float* my = red + t * 17;
#pragma unroll
    for (int p = 0; p < 4; ++p) {
        my[p * 4 + 0] = wy[p] * rq0; my[p * 4 + 1] = wy[p] * rq1;
        my[p * 4 + 2] = wy[p] * rq2; my[p * 4 + 3] = wy[p] * rq3;
    }
    __syncthreads();
    for (int s = 128; s > 0; s >>= 1) {
        if (t < s) {
#pragma unroll
            for (int k = 0; k < 16; ++k) red[t * 17 + k] += red[(t + s) * 17 + k];
        }
        __syncthreads();
    }
    if (t < 16) vproj[(size_t)blockIdx.x * 16 + t] = red[t];
}

// ---------------------------------------------------------------------------
// Kernel 2 (single block): vr[b,r] = (k1[r,:] . vproj[b,:]) / 16384
//                          m[b,o,pq] = sum_r k2[r,o,pq] * vr[b,r]
// ---------------------------------------------------------------------------
__global__ void small_kernel(const float* __restrict__ k1, const float* __restrict__ k2,
                             const float* __restrict__ vproj,
                             float* __restrict__ vr, float* __restrict__ m) {
    __shared__ float lvr[256];
    int t = threadIdx.x;
    {
        int b = t >> 5, r = t & 31;
        const float4* a = (const float4*)(k1 + (size_t)r * 2048);
        const float4* x = (const float4*)(vproj + (size_t)b * 2048);
        float acc = 0.f;
        for (int j = 0; j < 512; ++j) {
            float4 av = a[j], xv = x[j];
            acc += av.x * xv.x + av.y * xv.y + av.z * xv.z + av.w * xv.w;
        }
        acc *= (1.0f / 16384.0f);
        vr[t] = acc;
        lvr[t] = acc;
    }
    __syncthreads();
    for (int idx = t; idx < 16384; idx += 256) {
        int b   = idx >> 11;
        int rem = idx & 2047;      // o*16 + pq
        float acc = 0.f;
#pragma unroll 8
        for (int r = 0; r < 32; ++r)
            acc += k2[(size_t)r * 2048 + rem] * lvr[b * 32 + r];
        m[idx] = acc;
    }
}

// ---------------------------------------------------------------------------
// Kernel 3 (main): per block = one (b,h) row.
//   out[b,o,h,w] = WMMA( conv_w, v[b,:,h,:] ) + sum_q wx[w,q]*mh[o,q] + cbb[o]
//   mh[o,q] = sum_p wy[h,p]*m[b,o,p,q]  (computed once per block)
// 8 waves, wave ot owns Cout rows [16*ot,16*ot+16), loops 8 N-tiles of 16 pixels,
// K = Cin = 128 in 32 steps of V_WMMA_F32_16X16X4_F32 split over two independent
// accumulators (even/odd K-steps) to break the WMMA C-operand RAW chain.
// ---------------------------------------------------------------------------
__global__ void __launch_bounds__(256)
main_kernel(const float* __restrict__ v, const float* __restrict__ Aswz,
            const float* __restrict__ m, const float* __restrict__ cbb,
            float* __restrict__ out) {
    __shared__ float ldsV[CIN * LDSTRIDE];
    __shared__ float ldsMH[COUT * 4];

    int b = blockIdx.x >> 7;
    int h = blockIdx.x & 127;
    int t = threadIdx.x;

    // --- stage v[b, :, h, :] into LDS (i-major, padded stride) ---
    {
        int i = t >> 1, half = t & 1;
        const float4* g = (const float4*)(v + ((size_t)(b * CIN + i)) * (HWD * HWD) + h * HWD + half * 64);
        float4* l = (float4*)(ldsV + i * LDSTRIDE + half * 64);
#pragma unroll
        for (int j = 0; j < 16; ++j) l[j] = g[j];
    }
    // --- fold wy into m: mh[o][q] ---
    {
        float wy[4]; interp_w(h, wy);
        int o = t >> 1, q0 = (t & 1) * 2;
        const float* mb = m + (size_t)(b * COUT + o) * 16;
        float s0 = 0.f, s1 = 0.f;
#pragma unroll
        for (int p = 0; p < 4; ++p) {
            s0 += wy[p] * mb[p * 4 + q0];
            s1 += wy[p] * mb[p * 4 + q0 + 1];
        }
        ldsMH[o * 4 + q0]     = s0;
        ldsMH[o * 4 + q0 + 1] = s1;
    }
    __syncthreads();

    int wave = t >> 5, lane = t & 31;
    int n = lane & 15, halfl = lane >> 4;
    int kb = halfl << 1;
    int ot = wave;

    // --- preload A panel (pre-swizzled conv_w) for this Cout tile: 32 x v2f ---
    v2f a[32];
    {
        const v2f* ap = (const v2f*)Aswz + (size_t)ot * 1024 + lane;
#pragma unroll
        for (int ks = 0; ks < 32; ++ks) a[ks] = ap[ks * 32];
    }
    float cb[8];
#pragma unroll
    for (int j = 0; j < 8; ++j) cb[j] = cbb[ot * 16 + j + halfl * 8];

    float* obase = out + (size_t)b * (COUT * HWD * HWD) + h * HWD;

    for (int nt = 0; nt < 8; ++nt) {
        const float* bp = ldsV + nt * 16 + n;
        v8f acc0 = {}, acc1 = {};
#pragma unroll
        for (int ks = 0; ks < 32; ks += 2) {
            float b0 = bp[(ks * 4 + kb) * LDSTRIDE];
            float b1 = bp[(ks * 4 + kb + 1) * LDSTRIDE];
            float c0 = bp[((ks + 1) * 4 + kb) * LDSTRIDE];
            float c1 = bp[((ks + 1) * 4 + kb + 1) * LDSTRIDE];
            v2f bf0 = {b0, b1};
            v2f bf1 = {c0, c1};
            acc0 = __builtin_amdgcn_wmma_f32_16x16x4_f32(
                false, a[ks], false, bf0, (short)0, acc0, false, false);
            acc1 = __builtin_amdgcn_wmma_f32_16x16x4_f32(
                false, a[ks + 1], false, bf1, (short)0, acc1, false, false);
        }
        // --- fused epilogue: interp term + conv_b + bias ---
        int wpix = nt * 16 + n;
        float wx[4]; interp_w(wpix, wx);
#pragma unroll
        for (int j = 0; j < 8; ++j) {
            int o = ot * 16 + j + halfl * 8;
            float4 mh4 = *(const float4*)(ldsMH + o * 4);
            float t1 = wx[0] * mh4.x + wx[1] * mh4.y + wx[2] * mh4.z + wx[3] * mh4.w;
            obase[(size_t)o * (HWD * HWD) + wpix] = acc0[j] + acc1[j] + t1 + cb[j];
        }
    }
}

// ---------------------------------------------------------------------------
extern "C" void kernel_launch(void* const* d_in, const int* in_sizes, int n_in,
                              void* d_out, int out_size, void* d_ws, size_t ws_size,
                              hipStream_t stream) {
    const float* v      = (const float*)d_in[0];
    const float* k1     = (const float*)d_in[1];
    const float* k2     = (const float*)d_in[2];
    const float* conv_w = (const float*)d_in[3];
    const float* conv_b = (const float*)d_in[4];
    const float* bias   = (const float*)d_in[5];
    float* out = (float*)d_out;

    float* ws    = (float*)d_ws;
    float* vproj = ws;            // 16384 floats
    float* vr    = ws + 16384;    //   256
    float* m     = ws + 16640;    // 16384
    float* Aswz  = ws + 33024;    // 16384
    float* cbb   = ws + 49408;    //   128   (total ~198 KB)

    prep_kernel <<<33,   256, 0, stream>>>(conv_w, conv_b, bias, Aswz, cbb);
    vproj_kernel<<<1024, 256, 0, stream>>>(v, vproj);
    small_kernel<<<1,    256, 0, stream>>>(k1, k2, vproj, vr, m);
    main_kernel <<<1024, 256, 0, stream>>>(v, Aswz, m, cbb, out);
}